// DecodeLayer_74809740361968
// MI455X (gfx1250) — compile-verified
//
#include <hip/hip_runtime.h>

typedef __attribute__((ext_vector_type(2))) float v2f;
typedef __attribute__((ext_vector_type(8))) float v8f;

#define EMBED 4096
#define NUM_HEADS 32
#define HEAD_DIM 128
#define MAX_LEN 4096
#define BATCH 8
#define CHUNK 32          // K-dim chunk staged in LDS (double buffered)
#define SW (CHUNK + 4)    // padded LDS row stride (floats): conflict-free fragment reads
#define NCHUNK (EMBED / CHUNK)

// CDNA5 async copy: global -> LDS directly, tracked by ASYNCcnt (no VGPR bounce).
__device__ __forceinline__ void async_load_b128(unsigned lds_off, const float* gptr) {
    asm volatile("global_load_async_to_lds_b128 %0, %1, off"
                 :: "v"(lds_off), "v"(gptr)
                 : "memory");
}
__device__ __forceinline__ void wait_async0() {
    asm volatile("s_wait_asynccnt 0" ::: "memory");
}

// C(8 x 4096) = A(8 x 4096) @ W^T + bias, via V_WMMA_F32_16X16X4_F32.
// 8 waves per block, each wave owns one 16-wide N tile (16 rows of W).
// Double-buffered LDS; chunk c+1 streams in asynchronously while chunk c is computed.
// x buffer padded to 16 rows, rows 8..15 zeroed once -> unconditional ds_load_b64
// A-fragments (no exec-mask churn around the WMMAs, which need EXEC all-1s anyway).
__global__ __launch_bounds__(256) void gemm8_wmma(const float* __restrict__ A,
                                                  const float* __restrict__ W,
                                                  const float* __restrict__ bias,
                                                  float* __restrict__ out) {
    __shared__ float xs[2][16 * SW];         // x chunk (rows 0..7 data, 8..15 zero)
    __shared__ float wt[2][8][16 * SW];      // per-wave W tile (16 rows x CHUNK)

    const int tid  = threadIdx.x;
    const int wave = tid >> 5;
    const int lane = tid & 31;
    const int n0   = (blockIdx.x * 8 + wave) * 16;   // first W row / C column of tile
    const int mA   = lane & 15;                      // fragment row (A) / col (B)
    const int ko   = (lane >> 4) * 2;                // K sub-offset per fragment layout

    // zero the M-padding rows (8..15) of both x buffers, once
    for (int i = tid; i < 8 * SW; i += 256) {
        xs[0][8 * SW + i] = 0.0f;
        xs[1][8 * SW + i] = 0.0f;
    }

    auto issue_chunk = [&](int kbase, int buf) {
        // x chunk: 8 rows x 32 floats = 64 float4 -> threads 0..63
        if (tid < 64) {
            const int r   = tid >> 3;
            const int col = (tid & 7) << 2;
            async_load_b128((unsigned)(uintptr_t)&xs[buf][r * SW + col],
                            A + (size_t)r * EMBED + kbase + col);
        }
        // this wave's W tile: 16 rows x 32 floats, 4 rows per pass, coalesced 16B/lane
        const int col = (lane & 7) << 2;
        const int rb  = lane >> 3;               // 0..3
        #pragma unroll
        for (int it = 0; it < 4; ++it) {
            const int r = it * 4 + rb;
            async_load_b128((unsigned)(uintptr_t)&wt[buf][wave][r * SW + col],
                            W + (size_t)(n0 + r) * EMBED + kbase + col);
        }
    };

    v8f acc = {};
    issue_chunk(0, 0);

    for (int c = 0; c < NCHUNK; ++c) {
        wait_async0();        // this wave's chunk-c async copies are in LDS
        __syncthreads();      // everyone's chunk-c visible; prev buffer free
        if (c + 1 < NCHUNK) issue_chunk((c + 1) * CHUNK, (c + 1) & 1);

        const int buf = c & 1;
        const float* xrow = &xs[buf][mA * SW];
        const float* wrow = &wt[buf][wave][mA * SW];
        #pragma unroll
        for (int kk = 0; kk < CHUNK / 4; ++kk) {
            const int kc = kk * 4 + ko;
            const v2f a = *(const v2f*)(xrow + kc);   // ds_load_b64, rows >=8 are zero
            const v2f b = *(const v2f*)(wrow + kc);   // ds_load_b64
            acc = __builtin_amdgcn_wmma_f32_16x16x4_f32(false, a, false, b,
                                                        (short)0, acc, false, false);
        }
    }

    // C/D layout: lanes 0-15 hold M=0..7 in acc[0..7] at column N=n0+lane
    if (lane < 16) {
        const int col = n0 + lane;
        const float bv = bias[col];
        #pragma unroll
        for (int i = 0; i < 8; ++i)
            out[(size_t)i * EMBED + col] = acc[i] + bv;
    }
}

// One block per (b,h). Flash-style single pass over L = cur_len+1 positions.
// Position cur_len reads the freshly projected k/v from workspace (inputs untouched).
__global__ __launch_bounds__(256) void attn_decode(const float* __restrict__ q,
                                                   const float* __restrict__ k_cache,
                                                   const float* __restrict__ v_cache,
                                                   const float* __restrict__ k_new,
                                                   const float* __restrict__ v_new,
                                                   const int* __restrict__ cur_len_p,
                                                   float* __restrict__ attn_out) {
    const int bh   = blockIdx.x;         // b*NUM_HEADS + h
    const int b    = bh >> 5;
    const int h    = bh & 31;
    const int tid  = threadIdx.x;
    const int wave = tid >> 5;
    const int lane = tid & 31;
    const int cl   = *cur_len_p;

    const float scale = 0.088388347648318447f;  // 1/sqrt(128)
    float4 qf = *(const float4*)(q + (size_t)b * EMBED + h * HEAD_DIM + lane * 4);
    qf.x *= scale; qf.y *= scale; qf.z *= scale; qf.w *= scale;

    const size_t kvbase = (size_t)bh * MAX_LEN * HEAD_DIM;
    const float* kn = k_new + (size_t)b * EMBED + h * HEAD_DIM;
    const float* vn = v_new + (size_t)b * EMBED + h * HEAD_DIM;

    float m = -__builtin_inff();
    float l = 0.0f;
    float o0 = 0.0f, o1 = 0.0f, o2 = 0.0f, o3 = 0.0f;

    for (int p = wave; p <= cl; p += 8) {
        const bool isNew = (p == cl);
        const float* kr = isNew ? kn : (k_cache + kvbase + (size_t)p * HEAD_DIM);
        const float* vr = isNew ? vn : (v_cache + kvbase + (size_t)p * HEAD_DIM);

        const float4 kf = *(const float4*)(kr + lane * 4);
        float part = qf.x * kf.x + qf.y * kf.y + qf.z * kf.z + qf.w * kf.w;
        #pragma unroll
        for (int off = 16; off > 0; off >>= 1)
            part += __shfl_xor(part, off, 32);
        const float s = part;

        const float mn  = fmaxf(m, s);
        const float co  = __expf(m - mn);   // 0 when m == -inf
        const float wgt = __expf(s - mn);
        const float4 vf = *(const float4*)(vr + lane * 4);
        o0 = o0 * co + wgt * vf.x;
        o1 = o1 * co + wgt * vf.y;
        o2 = o2 * co + wgt * vf.z;
        o3 = o3 * co + wgt * vf.w;
        l = l * co + wgt;
        m = mn;
    }

    __shared__ float sm[8];
    __shared__ float sl[8];
    __shared__ float so[8][HEAD_DIM];
    if (lane == 0) { sm[wave] = m; sl[wave] = l; }
    so[wave][lane * 4 + 0] = o0;
    so[wave][lane * 4 + 1] = o1;
    so[wave][lane * 4 + 2] = o2;
    so[wave][lane * 4 + 3] = o3;
    __syncthreads();

    if (tid < HEAD_DIM) {
        float M = sm[0];
        #pragma unroll
        for (int w = 1; w < 8; ++w) M = fmaxf(M, sm[w]);
        float L = 0.0f, O = 0.0f;
        #pragma unroll
        for (int w = 0; w < 8; ++w) {
            const float cw = __expf(sm[w] - M);
            L += sl[w] * cw;
            O += so[w][tid] * cw;
        }
        attn_out[(size_t)b * EMBED + h * HEAD_DIM + tid] = O / L;
    }
}

extern "C" void kernel_launch(void* const* d_in, const int* in_sizes, int n_in,
                              void* d_out, int out_size, void* d_ws, size_t ws_size,
                              hipStream_t stream) {
    const float* x       = (const float*)d_in[0];
    const float* k_cache = (const float*)d_in[1];
    const float* v_cache = (const float*)d_in[2];
    const float* Wq      = (const float*)d_in[3];
    const float* bq      = (const float*)d_in[4];
    const float* Wk      = (const float*)d_in[5];
    const float* bk      = (const float*)d_in[6];
    const float* Wv      = (const float*)d_in[7];
    const float* bv      = (const float*)d_in[8];
    const float* Wo      = (const float*)d_in[9];
    const float* bo      = (const float*)d_in[10];
    const int*   cur_len = (const int*)d_in[11];
    float* out = (float*)d_out;

    float* q    = (float*)d_ws;                 // 8 x 4096
    float* kn   = q  + (size_t)BATCH * EMBED;   // 8 x 4096
    float* vn   = kn + (size_t)BATCH * EMBED;   // 8 x 4096
    float* attn = vn + (size_t)BATCH * EMBED;   // 8 x 4096

    gemm8_wmma<<<32, 256, 0, stream>>>(x, Wq, bq, q);
    gemm8_wmma<<<32, 256, 0, stream>>>(x, Wk, bk, kn);
    gemm8_wmma<<<32, 256, 0, stream>>>(x, Wv, bv, vn);
    attn_decode<<<BATCH * NUM_HEADS, 256, 0, stream>>>(q, k_cache, v_cache, kn, vn,
                                                       cur_len, attn);
    gemm8_wmma<<<32, 256, 0, stream>>>(attn, Wo, bo, out);
}